// SpecAR_Block_5016521801871
// MI455X (gfx1250) — compile-verified
//
#include <hip/hip_runtime.h>
#include <hip/hip_bf16.h>

// ---------------------------------------------------------------------------
// Problem constants
// ---------------------------------------------------------------------------
#define BATCH   64
#define TLEN    480
#define NCH     32
#define FDIM    64          // NFFT rows
#define NFFT    64
#define HOP     4
#define FRAMES  121
#define KP      128         // padded frame dimension (cols 121..127 are zero in A)

#define SST     132         // LDS row stride for S  (132 % 64 == 4 -> conflict free)
#define QST     68          // LDS row stride for QK (68 % 64 == 4)

typedef __attribute__((ext_vector_type(2))) float v2f;
typedef __attribute__((ext_vector_type(8))) float v8f;

__device__ __forceinline__ v8f wmma4(v2f a, v2f b, v8f c) {
    // D(16x16,f32) = A(16x4,f32) x B(4x16,f32) + C
    return __builtin_amdgcn_wmma_f32_16x16x4_f32(false, a, false, b, (short)0, c, false, false);
}

// ---------------------------------------------------------------------------
// Kernel 1: fused STFT (reflect pad + hann + DFT/8) and complex attention
//   S = stft(x[b,n,:])                      (64 x 128, cols>=121 zero)  in LDS
//   QK = sigmoid(S S^T complex)             (64 x 64)                   in LDS
//   A  = QK (complex-mul) S                 (64 x 128)                  -> global
// One workgroup (256 thr, 8 waves) per (b,n).
// ---------------------------------------------------------------------------
__global__ __launch_bounds__(256) void k1_stft_attn(
        const float* __restrict__ x, float* __restrict__ Ar, float* __restrict__ Ai)
{
    extern __shared__ float sm[];
    float* xrow = sm;                 // 512
    float* win  = sm + 512;           // 64
    float* twc  = sm + 576;           // 64
    float* tws  = sm + 640;           // 64
    float* s_r  = sm + 704;           // FDIM*SST
    float* s_i  = s_r + FDIM * SST;
    float* qk_r = s_i + FDIM * SST;   // FDIM*QST
    float* qk_i = qk_r + FDIM * QST;

    const int bn = blockIdx.x;
    const int b  = bn >> 5, n = bn & 31;
    const int tid = threadIdx.x;

    for (int t = tid; t < TLEN; t += 256)
        xrow[t] = x[(size_t)b * TLEN * NCH + (size_t)t * NCH + n];
    if (tid < 64) {
        float ang = 6.283185307179586f * (float)tid / 64.0f;
        float c = cosf(ang), s = sinf(ang);
        twc[tid] = c; tws[tid] = s;
        win[tid] = 0.5f * (1.0f - c);
    }
    __syncthreads();

    // ---- STFT: S[f, frame] ----
    for (int o = tid; o < FDIM * FRAMES; o += 256) {
        int f  = o / FRAMES;
        int fr = o - f * FRAMES;
        int base = fr * HOP - (NFFT / 2);
        float ar = 0.f, ai = 0.f;
        #pragma unroll 4
        for (int t = 0; t < NFFT; ++t) {
            int q = base + t;                              // reflect padding
            q = (q < 0) ? -q : ((q > TLEN - 1) ? (2 * (TLEN - 1) - q) : q);
            float v = xrow[q] * win[t];
            int m = (f * t) & 63;
            ar += v * twc[m];
            ai -= v * tws[m];
        }
        s_r[f * SST + fr] = ar * 0.125f;   // 1/sqrt(64)
        s_i[f * SST + fr] = ai * 0.125f;
    }
    for (int o = tid; o < FDIM * (KP - FRAMES); o += 256) {  // zero tail cols
        int f = o / (KP - FRAMES);
        int c = FRAMES + (o - f * (KP - FRAMES));
        s_r[f * SST + c] = 0.f;
        s_i[f * SST + c] = 0.f;
    }
    __syncthreads();

    const int wv = tid >> 5, lane = tid & 31;
    const int l15 = lane & 15, lh = lane >> 4;

    // ---- QK = sigmoid(S S^T complex): 16 C-tiles (4x4), 2 per wave ----
    #pragma unroll
    for (int tt = 0; tt < 2; ++tt) {
        int ti = wv * 2 + tt;
        int f0 = (ti >> 2) * 16, g0 = (ti & 3) * 16;
        v8f accRR = {}, accII = {}, accI = {};
        for (int k0 = 0; k0 < KP; k0 += 4) {
            int ka = k0 + 2 * lh;
            const float* pf_r = &s_r[(f0 + l15) * SST + ka];
            const float* pf_i = &s_i[(f0 + l15) * SST + ka];
            const float* pg_r = &s_r[(g0 + l15) * SST + ka];
            const float* pg_i = &s_i[(g0 + l15) * SST + ka];
            v2f Afr = { pf_r[0], pf_r[1] };
            v2f Afi = { pf_i[0], pf_i[1] };
            v2f Bgr = { pg_r[0], pg_r[1] };
            v2f Bgi = { pg_i[0], pg_i[1] };
            accRR = wmma4(Afr, Bgr, accRR);
            accII = wmma4(Afi, Bgi, accII);
            accI  = wmma4(Afr, Bgi, accI);
            accI  = wmma4(Afi, Bgr, accI);
        }
        #pragma unroll
        for (int j = 0; j < 8; ++j) {
            int row = f0 + j + 8 * lh, col = g0 + l15;
            float vr = accRR[j] - accII[j];
            float vi = accI[j];
            qk_r[row * QST + col] = 1.0f / (1.0f + __expf(-vr));
            qk_i[row * QST + col] = 1.0f / (1.0f + __expf(-vi));
        }
    }
    __syncthreads();

    // ---- A = QK (complex) S : 32 C-tiles (4 f-tiles x 8 k-tiles), 4 per wave ----
    const size_t obase = (size_t)bn * FDIM * KP;
    #pragma unroll
    for (int tt = 0; tt < 4; ++tt) {
        int ti = wv * 4 + tt;
        int f0 = (ti >> 3) * 16, kb = (ti & 7) * 16;
        v8f aP = {}, aM = {}, aI = {};
        for (int g0 = 0; g0 < FDIM; g0 += 4) {
            int qa = (f0 + l15) * QST + g0 + 2 * lh;
            v2f Qr = { qk_r[qa], qk_r[qa + 1] };
            v2f Qi = { qk_i[qa], qk_i[qa + 1] };
            int sa = (g0 + 2 * lh) * SST + kb + l15;
            v2f Sr = { s_r[sa], s_r[sa + SST] };
            v2f Si = { s_i[sa], s_i[sa + SST] };
            aP = wmma4(Qr, Sr, aP);
            aM = wmma4(Qi, Si, aM);
            aI = wmma4(Qr, Si, aI);
            aI = wmma4(Qi, Sr, aI);
        }
        #pragma unroll
        for (int j = 0; j < 8; ++j) {
            int row = f0 + j + 8 * lh, col = kb + l15;
            Ar[obase + (size_t)row * KP + col] = aP[j] - aM[j];
            Ai[obase + (size_t)row * KP + col] = aI[j];
        }
    }
}

// ---------------------------------------------------------------------------
// Kernel 2: generic complex conv (implicit GEMM on WMMA f32 16x16x4)
//   out_r = conv(in_r, Wr) - conv(in_i, Wi) + (br - bi)
//   out_i = conv(in_i, Wr) + conv(in_r, Wi) + (br + bi)
//   taps: 1 (1x1) or 9 (3x3, dilation dil). mode: 0=write, 1=relu-write,
//   2=relu-accumulate. Grid: BATCH*8 blocks (8 f-rows each), 256 thr.
// ---------------------------------------------------------------------------
__global__ __launch_bounds__(256) void k_conv_complex(
        const float* __restrict__ inR, const float* __restrict__ inI,
        const float* __restrict__ Wr,  const float* __restrict__ Wi,
        const float* __restrict__ br,  const float* __restrict__ bi,
        float* __restrict__ outR, float* __restrict__ outI,
        int taps, int dil, int mode)
{
    extern __shared__ float sw[];
    const int wst = taps * NCH + 1;          // padded per-co stride (odd, conflict free)
    float* wr_l = sw;                        // NCH * wst
    float* wi_l = sw + NCH * wst;

    const int tid = threadIdx.x;
    const int nW = NCH * NCH * taps;
    for (int i = tid; i < nW; i += 256) {
        int co = i / (NCH * taps);
        int r  = i - co * (NCH * taps);
        wr_l[co * wst + r] = Wr[i];
        wi_l[co * wst + r] = Wi[i];
    }
    __syncthreads();

    const int b  = blockIdx.x >> 3;
    const int fq = blockIdx.x & 7;
    const int wv = tid >> 5, lane = tid & 31;
    const int l15 = lane & 15, lh = lane >> 4;
    const int kb = wv * 16;
    const size_t inBase  = (size_t)b * NCH * FDIM * KP;
    const size_t chPitch = (size_t)FDIM * KP;

    for (int it = 0; it < 8; ++it) {
        int f = fq * 8 + it;
        v8f acc[8];
        #pragma unroll
        for (int q = 0; q < 8; ++q) acc[q] = (v8f){};

        for (int tap = 0; tap < taps; ++tap) {
            int df = (taps == 1) ? 0 : ((tap / 3) - 1) * dil;
            int dk = (taps == 1) ? 0 : ((tap % 3) - 1) * dil;
            int fr = f + df;
            int c  = kb + l15 + dk;
            bool ok = (fr >= 0) && (fr < FDIM) && (c >= 0) && (c < FRAMES);
            size_t pbase = inBase + (size_t)fr * KP + (size_t)c;
            for (int ci0 = 0; ci0 < NCH; ci0 += 4) {
                int ciA = ci0 + 2 * lh;
                float xr0 = 0.f, xr1 = 0.f, xi0 = 0.f, xi1 = 0.f;
                if (ok) {
                    size_t p0 = pbase + (size_t)ciA * chPitch;
                    size_t p1 = p0 + chPitch;
                    xr0 = inR[p0]; xr1 = inR[p1];
                    xi0 = inI[p0]; xi1 = inI[p1];
                }
                v2f bR = { xr0, xr1 };
                v2f bI = { xi0, xi1 };
                int w0 = l15 * wst + ciA * taps + tap;         // co-tile 0
                int w1 = (16 + l15) * wst + ciA * taps + tap;  // co-tile 1
                v2f aR0 = { wr_l[w0], wr_l[w0 + taps] };
                v2f aI0 = { wi_l[w0], wi_l[w0 + taps] };
                v2f aR1 = { wr_l[w1], wr_l[w1 + taps] };
                v2f aI1 = { wi_l[w1], wi_l[w1 + taps] };
                acc[0] = wmma4(aR0, bR, acc[0]);   // RR tile0
                acc[1] = wmma4(aI0, bI, acc[1]);   // II tile0
                acc[2] = wmma4(aR0, bI, acc[2]);   // RI tile0
                acc[3] = wmma4(aI0, bR, acc[3]);   // IR tile0
                acc[4] = wmma4(aR1, bR, acc[4]);
                acc[5] = wmma4(aI1, bI, acc[5]);
                acc[6] = wmma4(aR1, bI, acc[6]);
                acc[7] = wmma4(aI1, bR, acc[7]);
            }
        }

        #pragma unroll
        for (int ct = 0; ct < 2; ++ct) {
            #pragma unroll
            for (int j = 0; j < 8; ++j) {
                int row = ct * 16 + j + 8 * lh;
                int col = kb + l15;
                float bre = br[row], bie = bi[row];
                float vr = acc[ct * 4 + 0][j] - acc[ct * 4 + 1][j] + (bre - bie);
                float vi = acc[ct * 4 + 2][j] + acc[ct * 4 + 3][j] + (bre + bie);
                size_t o = inBase + (size_t)row * chPitch + (size_t)f * KP + col;
                if (mode == 0)       { outR[o] = vr;                 outI[o] = vi; }
                else if (mode == 1)  { outR[o] = fmaxf(vr, 0.f);     outI[o] = fmaxf(vi, 0.f); }
                else                 { outR[o] += fmaxf(vr, 0.f);    outI[o] += fmaxf(vi, 0.f); }
            }
        }
    }
}

// ---------------------------------------------------------------------------
// Kernel 3: l0(2->1) + mean over f:   h[b,n,k] = sum_f(Mr)*w00/(3*64)
//                                              + sum_f(Mi)*w01/(3*64) + b_l0
// ---------------------------------------------------------------------------
__global__ void k5_reduce(const float* __restrict__ Mr, const float* __restrict__ Mi,
                          const float* __restrict__ w_l0, const float* __restrict__ b_l0,
                          float* __restrict__ h)
{
    int idx = blockIdx.x * 256 + threadIdx.x;
    if (idx >= BATCH * NCH * FRAMES) return;
    int k  = idx % FRAMES;
    int bn = idx / FRAMES;
    size_t base = (size_t)bn * FDIM * KP + k;
    float sr = 0.f, si = 0.f;
    #pragma unroll 4
    for (int f = 0; f < FDIM; ++f) {
        sr += Mr[base + (size_t)f * KP];
        si += Mi[base + (size_t)f * KP];
    }
    const float sc = 1.0f / (3.0f * 64.0f);
    h[(size_t)bn * FRAMES + k] = sr * sc * w_l0[0] + si * sc * w_l0[1] + b_l0[0];
}

// ---------------------------------------------------------------------------
// Kernel 4: l2 (FRAMES -> T) + transpose to [B, T, N]
// ---------------------------------------------------------------------------
__global__ void k6_out(const float* __restrict__ h, const float* __restrict__ w_l2,
                       const float* __restrict__ b_l2, float* __restrict__ out)
{
    int idx = blockIdx.x * 256 + threadIdx.x;
    if (idx >= BATCH * TLEN * NCH) return;
    int n = idx % NCH;
    int t = (idx / NCH) % TLEN;
    int b = idx / (NCH * TLEN);
    const float* hp = h + ((size_t)b * NCH + n) * FRAMES;
    const float* wp = w_l2 + (size_t)t * FRAMES;
    float acc = b_l2[t];
    #pragma unroll 4
    for (int k = 0; k < FRAMES; ++k) acc += hp[k] * wp[k];
    out[idx] = acc;
}

// ---------------------------------------------------------------------------
// Launcher
// ---------------------------------------------------------------------------
extern "C" void kernel_launch(void* const* d_in, const int* in_sizes, int n_in,
                              void* d_out, int out_size, void* d_ws, size_t ws_size,
                              hipStream_t stream) {
    (void)in_sizes; (void)n_in; (void)out_size; (void)ws_size;

    const float* x    = (const float*)d_in[0];
    const float* w0r  = (const float*)d_in[1];
    const float* w0i  = (const float*)d_in[2];
    const float* b0r  = (const float*)d_in[3];
    const float* b0i  = (const float*)d_in[4];
    const float* W1r  = (const float*)d_in[5];
    const float* W1i  = (const float*)d_in[6];
    const float* B1r  = (const float*)d_in[7];
    const float* B1i  = (const float*)d_in[8];
    const float* W2r  = (const float*)d_in[9];
    const float* W2i  = (const float*)d_in[10];
    const float* B2r  = (const float*)d_in[11];
    const float* B2i  = (const float*)d_in[12];
    const float* w_l0 = (const float*)d_in[13];
    const float* b_l0 = (const float*)d_in[14];
    const float* w_l2 = (const float*)d_in[15];
    const float* b_l2 = (const float*)d_in[16];
    float* out = (float*)d_out;

    float* ws = (float*)d_ws;
    const size_t NA = (size_t)BATCH * NCH * FDIM * KP;   // 16.78M floats per plane
    float* Ar = ws;
    float* Ai = Ar + NA;
    float* Mr = Ai + NA;
    float* Mi = Mr + NA;
    float* Tr = Mi + NA;
    float* Ti = Tr + NA;
    float* h  = Ti + NA;

    // 1) fused STFT + attention -> A
    const int SM1 = (704 + 2 * FDIM * SST + 2 * FDIM * QST) * (int)sizeof(float);
    k1_stft_attn<<<BATCH * NCH, 256, SM1, stream>>>(x, Ar, Ai);

    const int SMC1 = (2 * NCH * (NCH * 1 + 1)) * (int)sizeof(float);
    const int SMC9 = (2 * NCH * (NCH * 9 + 1)) * (int)sizeof(float);

    // 2) 1x1 complex conv, relu, initialize M
    k_conv_complex<<<BATCH * 8, 256, SMC1, stream>>>(
        Ar, Ai, w0r, w0i, b0r, b0i, Mr, Mi, 1, 1, 1);

    // 3) dilated 3x3 complex conv pairs, relu-accumulate into M
    for (int k = 1; k <= 2; ++k) {
        int j = k - 1;
        size_t wo = (size_t)j * NCH * NCH * 9;
        size_t bo = (size_t)j * NCH;
        k_conv_complex<<<BATCH * 8, 256, SMC9, stream>>>(
            Ar, Ai, W1r + wo, W1i + wo, B1r + bo, B1i + bo, Tr, Ti, 9, k, 0);
        k_conv_complex<<<BATCH * 8, 256, SMC9, stream>>>(
            Tr, Ti, W2r + wo, W2i + wo, B2r + bo, B2i + bo, Mr, Mi, 9, k, 2);
    }

    // 4) l0 + mean over f
    const int nh = BATCH * NCH * FRAMES;
    k5_reduce<<<(nh + 255) / 256, 256, 0, stream>>>(Mr, Mi, w_l0, b_l0, h);

    // 5) l2 + transpose
    const int no = BATCH * TLEN * NCH;
    k6_out<<<(no + 255) / 256, 256, 0, stream>>>(h, w_l2, b_l2, out);
}